// DynamicRouting2d_83983790506263
// MI455X (gfx1250) — compile-verified
//
#include <hip/hip_runtime.h>

typedef float v2f __attribute__((ext_vector_type(2)));
typedef float v8f __attribute__((ext_vector_type(8)));

// Capsule dynamic routing, fully fused.
// A=32 in-caps, B=32 out-caps, C=4 pose-in, D=4 pose-out, k=1, pad=1.
// pose: [8,128,30,30] fp32; W: [32,128,4] fp32; out: [8,128,32,32] fp32.
// Grid: 512 blocks (8 batch * 32 rows * 2 col-tiles), 512 threads (16 waves).
// Each block: 16 positions. Phase 1: votes[16][32][128] in LDS via
// v_wmma_f32_16x16x4_f32 (16 pos x 16 outdims, K=4). Phase 2: one wave per
// position, lane = output capsule B, 4 routing iterations.
__global__ __launch_bounds__(512) void caps_routing_kernel(
    const float* __restrict__ pose,
    const float* __restrict__ Wm,
    float* __restrict__ out)
{
    extern __shared__ float votes[];   // [16 pos][32 k][128 p] = 256 KB

    const int tid  = threadIdx.x;
    const int wave = tid >> 5;
    const int lane = tid & 31;

    const int blk = blockIdx.x;        // 0..511
    const int b   = blk >> 6;          // batch 0..7
    const int rem = blk & 63;
    const int y   = rem >> 1;          // output row 0..31
    const int x0  = (rem & 1) << 4;    // column tile base: 0 or 16

    // ---------------- Phase 1: votes via V_WMMA_F32_16X16X4_F32 -------------
    // A (16x4): lanes 0-15 -> M=lane, c={0,1}; lanes 16-31 -> M=lane-16, c={2,3}
    // B (4x16): lanes 0-15 -> N=lane, K={0,1}; lanes 16-31 -> N=lane-16, K={2,3}
    const int mh = lane & 15;          // position-in-tile (A) / column (B)
    const int hi = lane >> 4;
    const int cb = hi << 1;            // pose-dim base: 0 or 2
    const int x  = x0 + mh;
    // unfold(k=1,pad=1): position (y,x) reads input pixel (y-1,x-1), 0 if OOB
    const bool valid = (y >= 1) && (y <= 30) && (x >= 1) && (x <= 30);
    const long pbase = (long)b * 128 * 900 + (long)(y - 1) * 30 + (x - 1);

    for (int kk = 0; kk < 2; ++kk) {
        const int k  = (wave << 1) + kk;   // input capsule 0..31
        const int ch = (k << 2) + cb;      // pose channel a*4 + c
        v2f a;
        a.x = valid ? pose[pbase + (long)ch * 900]       : 0.0f;
        a.y = valid ? pose[pbase + (long)(ch + 1) * 900] : 0.0f;
        #pragma unroll
        for (int t = 0; t < 8; ++t) {
            const int col = (t << 4) + mh;                 // output dim 0..127
            const float* wp = Wm + (((k << 7) + col) << 2) + cb;  // W[k,col,c]
            v2f bm;
            bm.x = wp[0];
            bm.y = wp[1];
            v8f acc = {};
            acc = __builtin_amdgcn_wmma_f32_16x16x4_f32(
                      false, a, false, bm, (short)0, acc, false, false);
            // D layout: VGPR r, lanes 0-15 -> (M=r, N=lane); 16-31 -> (M=r+8)
            #pragma unroll
            for (int r = 0; r < 8; ++r) {
                const int m = hi ? (r + 8) : r;
                votes[((m << 5) + k) * 128 + col] = acc[r];
            }
        }
    }
    __syncthreads();

    // ---------------- Phase 2: dynamic routing (wave = position, lane = B) --
    const float* vp = votes + (wave << 5) * 128;   // [k][128]

    float b_log[32];
    #pragma unroll
    for (int k = 0; k < 32; ++k) b_log[k] = 0.0f;
    float v0 = 0.f, v1 = 0.f, v2 = 0.f, v3 = 0.f;

    for (int it = 0; it < 4; ++it) {               // ITERS+1 = 4
        // c[k][B] = softmax over B (cross-lane) of b_log[k][B]
        float cc[32];
        #pragma unroll
        for (int k = 0; k < 32; ++k) {
            float m = b_log[k];
            #pragma unroll
            for (int s = 16; s >= 1; s >>= 1)
                m = fmaxf(m, __shfl_xor(m, s, 32));
            const float e = __expf(b_log[k] - m);
            float Z = e;
            #pragma unroll
            for (int s = 16; s >= 1; s >>= 1)
                Z += __shfl_xor(Z, s, 32);
            cc[k] = e / Z;
        }
        // s[B][D] = sum_k c[k][B] * votes[k][B][D]   (lane-local over D)
        float s0 = 0.f, s1 = 0.f, s2 = 0.f, s3 = 0.f;
        #pragma unroll
        for (int k = 0; k < 32; ++k) {
            const float4 vv = *(const float4*)(vp + (k << 7) + (lane << 2));
            s0 = fmaf(cc[k], vv.x, s0);
            s1 = fmaf(cc[k], vv.y, s1);
            s2 = fmaf(cc[k], vv.z, s2);
            s3 = fmaf(cc[k], vv.w, s3);
        }
        // squash: (|s|^2/(1+|s|^2)) * s/sqrt(|s|^2+eps)
        const float n2 = s0 * s0 + s1 * s1 + s2 * s2 + s3 * s3;
        const float sc = (n2 / (1.0f + n2)) * __frsqrt_rn(n2 + 1e-8f);
        v0 = sc * s0; v1 = sc * s1; v2 = sc * s2; v3 = sc * s3;
        // b_log[k][B] += <v[B], votes[k][B]>  (skip on final iter: v is output)
        if (it < 3) {
            #pragma unroll
            for (int k = 0; k < 32; ++k) {
                const float4 vv = *(const float4*)(vp + (k << 7) + (lane << 2));
                b_log[k] += v0 * vv.x + v1 * vv.y + v2 * vv.z + v3 * vv.w;
            }
        }
    }

    // out[b, B*4+d, y, x], x = x0 + wave
    const long obase = ((long)b * 128 + (lane << 2)) * 1024 + y * 32 + (x0 + wave);
    out[obase]        = v0;
    out[obase + 1024] = v1;
    out[obase + 2048] = v2;
    out[obase + 3072] = v3;
}

extern "C" void kernel_launch(void* const* d_in, const int* in_sizes, int n_in,
                              void* d_out, int out_size, void* d_ws, size_t ws_size,
                              hipStream_t stream) {
    const float* pose = (const float*)d_in[0];   // [8,128,30,30]
    const float* Wm   = (const float*)d_in[1];   // [32,128,4]
    float* out        = (float*)d_out;           // [8,128,32,32]

    const size_t shmem = 16 * 32 * 128 * sizeof(float);   // 256 KB LDS
    (void)hipFuncSetAttribute((const void*)caps_routing_kernel,
                              hipFuncAttributeMaxDynamicSharedMemorySize,
                              (int)shmem);
    caps_routing_kernel<<<512, 512, shmem, stream>>>(pose, Wm, out);
}